// Proto_37125697307126
// MI455X (gfx1250) — compile-verified
//
#include <hip/hip_runtime.h>
#include <stdint.h>

#define B_    2
#define DIM_  256
#define C_    19
#define TOPK_ 256
#define HF_   128
#define WF_   128
#define H_    512
#define W_    512
#define N_    (B_*H_*W_)
#define GAMMA_ 0.999f
#define EPS_   1e-12f
#define HB1 65536   // level-1 bins: bits[30:15]
#define HB2 32768   // level-2 bins: bits[14:0]

typedef float v2f __attribute__((ext_vector_type(2)));
typedef float v8f __attribute__((ext_vector_type(8)));

static __device__ __forceinline__ unsigned int ballot32(bool p) {
#if __has_builtin(__builtin_amdgcn_ballot_w32)
  return __builtin_amdgcn_ballot_w32(p);
#else
  return (unsigned int)__ballot(p);
#endif
}

// ---------------------------------------------------------------------------
// K1: per-pixel upsampled weight for the pixel's own label class.
// jax.image.resize bilinear (half-pixel, edge clamp): src = (dst+0.5)*0.25-0.5
// ---------------------------------------------------------------------------
__global__ void k_weight_up(const float* __restrict__ weight,
                            const int* __restrict__ labels,
                            float* __restrict__ w_arr) {
  int n = blockIdx.x * blockDim.x + threadIdx.x;
  if (n >= N_) return;
  int lab = labels[n];
  float wv = 0.f;
  if (lab >= 0 && lab < C_) {
    int b = n >> 18;                 // / (512*512)
    int r = n & (H_*W_ - 1);
    int y = r >> 9, x = r & (W_ - 1);
    float fy = (y + 0.5f) * 0.25f - 0.5f;
    float fx = (x + 0.5f) * 0.25f - 0.5f;
    float fyf = floorf(fy), fxf = floorf(fx);
    float ty = fy - fyf, tx = fx - fxf;
    int y0 = (int)fyf, x0 = (int)fxf;
    int y1 = min(max(y0 + 1, 0), HF_ - 1), x1 = min(max(x0 + 1, 0), WF_ - 1);
    y0 = min(max(y0, 0), HF_ - 1);   x0 = min(max(x0, 0), WF_ - 1);
    const float* wp = weight + (size_t)(b * C_ + lab) * (HF_ * WF_);
    float w00 = wp[y0*WF_ + x0], w01 = wp[y0*WF_ + x1];
    float w10 = wp[y1*WF_ + x0], w11 = wp[y1*WF_ + x1];
    wv = (1.f-ty)*((1.f-tx)*w00 + tx*w01) + ty*((1.f-tx)*w10 + tx*w11);
    if (wv < 0.f) wv = 0.f;          // keep bit-pattern order == value order
  }
  w_arr[n] = wv;
}

// ---------------------------------------------------------------------------
// K2: per-class exact top-K threshold via two-level radix select.
// One workgroup per class; 256KB+4KB dynamic LDS histogram (CDNA5 WGP LDS).
// Outputs: thr bits (31-bit pattern), #ties-to-take, min(K, class_count).
// ---------------------------------------------------------------------------
__global__ void k_class_threshold(const float* __restrict__ w_arr,
                                  const int* __restrict__ labels,
                                  uint32_t* __restrict__ cls_thr,
                                  uint32_t* __restrict__ cls_need,
                                  uint32_t* __restrict__ cls_nsel) {
  extern __shared__ uint32_t lds[];        // HB1 bins + 1024 chunk sums
  uint32_t* hist = lds;
  uint32_t* csum = lds + HB1;
  __shared__ uint32_t sT, sNeedT, sDone;
  const int c = blockIdx.x;
  const int tid = threadIdx.x;
  const int NT = blockDim.x;               // 1024

  for (int i = tid; i < HB1; i += NT) hist[i] = 0;
  __syncthreads();
  for (int n = tid; n < N_; n += NT) {
    if (n + 8*NT < N_) __builtin_prefetch(&labels[n + 8*NT], 0, 1);
    if (labels[n] == c) {
      uint32_t bits = __float_as_uint(w_arr[n]);
      atomicAdd(&hist[bits >> 15], 1u);
    }
  }
  __syncthreads();
  { uint32_t s = 0; int base = tid * 64;
    for (int j = 0; j < 64; ++j) s += hist[base + j];
    csum[tid] = s; }
  __syncthreads();
  if (tid == 0) {
    uint32_t total = 0;
    for (int t = 0; t < NT; ++t) total += csum[t];
    if (total == 0) {
      cls_thr[c] = 0xFFFFFFFFu; cls_need[c] = 0; cls_nsel[c] = 0; sDone = 1;
    } else if (total <= TOPK_) {
      cls_thr[c] = 0u; cls_need[c] = TOPK_; cls_nsel[c] = total; sDone = 1;
    } else {
      uint32_t acc = 0; int T = 0;
      for (int t = NT - 1; t >= 0; --t) {
        if (acc + csum[t] >= TOPK_) {
          for (int b = t*64 + 63; b >= t*64; --b) {
            acc += hist[b];
            if (acc >= TOPK_) { T = b; break; }
          }
          break;
        }
        acc += csum[t];
      }
      sT = (uint32_t)T; sNeedT = TOPK_ - (acc - hist[T]);
      cls_nsel[c] = TOPK_; sDone = 0;
    }
  }
  __syncthreads();
  if (sDone) return;
  const uint32_t T = sT, needT = sNeedT;

  for (int i = tid; i < HB2; i += NT) hist[i] = 0;
  __syncthreads();
  for (int n = tid; n < N_; n += NT) {
    if (labels[n] == c) {
      uint32_t bits = __float_as_uint(w_arr[n]);
      if ((bits >> 15) == T) atomicAdd(&hist[bits & 0x7FFFu], 1u);
    }
  }
  __syncthreads();
  { uint32_t s = 0; int base = tid * 32;
    for (int j = 0; j < 32; ++j) s += hist[base + j];
    csum[tid] = s; }
  __syncthreads();
  if (tid == 0) {
    uint32_t acc = 0; int T2 = 0;
    for (int t = NT - 1; t >= 0; --t) {
      if (acc + csum[t] >= needT) {
        for (int b = t*32 + 31; b >= t*32; --b) {
          acc += hist[b];
          if (acc >= needT) { T2 = b; break; }
        }
        break;
      }
      acc += csum[t];
    }
    cls_thr[c]  = (T << 15) | (uint32_t)T2;
    cls_need[c] = needT - (acc - hist[T2]);
  }
}

// ---------------------------------------------------------------------------
// K3: deterministic index-ordered selection (matches jax top_k tie-break).
// One workgroup/class, 256 threads = 8 wave32s; ballot + popc prefix scan.
// ---------------------------------------------------------------------------
__global__ void k_select(const float* __restrict__ w_arr,
                         const int* __restrict__ labels,
                         const uint32_t* __restrict__ cls_thr,
                         const uint32_t* __restrict__ cls_need,
                         uint32_t* __restrict__ sel_idx,
                         float* __restrict__ sel_w) {
  const int c = blockIdx.x;
  const int tid = threadIdx.x;
  const int lane = tid & 31, wid = tid >> 5;
  __shared__ uint32_t wSel[8], wTie[8];
  __shared__ uint32_t selBase, tieBase;
  if (tid < TOPK_) { sel_idx[c*TOPK_ + tid] = 0u; sel_w[c*TOPK_ + tid] = 0.f; }
  if (tid == 0) { selBase = 0; tieBase = 0; }
  const uint32_t thr = cls_thr[c];
  const uint32_t needx = cls_need[c];
  __syncthreads();
  const uint32_t laneMaskLt = (lane == 0) ? 0u : ((1u << lane) - 1u);
  for (int n0 = 0; n0 < N_; n0 += 256) {
    int n = n0 + tid;
    bool isC = (labels[n] == c);
    uint32_t bits = isC ? __float_as_uint(w_arr[n]) : 0u;
    bool tie = isC && (bits == thr);
    uint32_t tmask = ballot32(tie);
    if (lane == 0) wTie[wid] = __popc(tmask);
    __syncthreads();
    uint32_t tieOff = tieBase;
    for (int w = 0; w < wid; ++w) tieOff += wTie[w];
    uint32_t myTieRank = tieOff + __popc(tmask & laneMaskLt);
    bool sel = isC && ((bits > thr) || (tie && myTieRank < needx));
    uint32_t smask = ballot32(sel);
    if (lane == 0) wSel[wid] = __popc(smask);
    __syncthreads();
    uint32_t selOff = selBase;
    for (int w = 0; w < wid; ++w) selOff += wSel[w];
    uint32_t slot = selOff + __popc(smask & laneMaskLt);
    if (sel && slot < TOPK_) {
      sel_idx[c*TOPK_ + slot] = (uint32_t)n;
      sel_w[c*TOPK_ + slot]   = __uint_as_float(bits);
    }
    __syncthreads();
    if (tid == 0) {
      uint32_t ts = 0, ss = 0;
      for (int w = 0; w < 8; ++w) { ts += wTie[w]; ss += wSel[w]; }
      tieBase += ts; selBase += ss;
    }
    __syncthreads();
  }
}

// ---------------------------------------------------------------------------
// K4: proto[c,:] = Sum_k w_k * bilinear_feat(idx_k,:) via V_WMMA_F32_16X16X4_F32,
// then /wsum, l2norm, EMA, final l2norm. One workgroup/class, 16 waves,
// each wave owns a 16-dim tile; A = weights broadcast, B = gathered features.
// ---------------------------------------------------------------------------
__global__ void k_proto(const float* __restrict__ features,
                        const uint32_t* __restrict__ sel_idx,
                        const float* __restrict__ sel_w,
                        const uint32_t* __restrict__ cls_nsel,
                        const float* __restrict__ prototypes,
                        const float* __restrict__ update_count,
                        float* __restrict__ out) {
  const int c = blockIdx.x;
  const int tid = threadIdx.x;          // 512 threads = 16 wave32s
  const int lane = tid & 31, wid = tid >> 5;
  __shared__ int   sBase[TOPK_][4];
  __shared__ float sCoef[TOPK_][4];
  __shared__ float sW[TOPK_];
  __shared__ float sProto[DIM_];
  __shared__ float sScalar[4];

  if (tid < TOPK_) {
    int k = tid;
    int n = (int)sel_idx[c*TOPK_ + k];
    float wv = sel_w[c*TOPK_ + k];
    int b = n >> 18;
    int r = n & (H_*W_ - 1);
    int y = r >> 9, x = r & (W_ - 1);
    float fy = (y + 0.5f) * 0.25f - 0.5f;
    float fx = (x + 0.5f) * 0.25f - 0.5f;
    float fyf = floorf(fy), fxf = floorf(fx);
    float ty = fy - fyf, tx = fx - fxf;
    int y0 = (int)fyf, x0 = (int)fxf;
    int y1 = min(max(y0 + 1, 0), HF_ - 1), x1 = min(max(x0 + 1, 0), WF_ - 1);
    y0 = min(max(y0, 0), HF_ - 1);       x0 = min(max(x0, 0), WF_ - 1);
    int base = b * (DIM_ * HF_ * WF_);
    sBase[k][0] = base + y0*WF_ + x0;  sBase[k][1] = base + y0*WF_ + x1;
    sBase[k][2] = base + y1*WF_ + x0;  sBase[k][3] = base + y1*WF_ + x1;
    sCoef[k][0] = (1.f-ty)*(1.f-tx);   sCoef[k][1] = (1.f-ty)*tx;
    sCoef[k][2] = ty*(1.f-tx);         sCoef[k][3] = ty*tx;
    sW[k] = wv;
  }
  __syncthreads();

  // WMMA A(16x4) lane layout: lanes<16 -> K={0,1}, lanes>=16 -> K={2,3}
  // WMMA B(4x16):  lanes<16 -> N=lane,K={0,1}; lanes>=16 -> N=lane-16,K={2,3}
  const int d0  = wid * 16;
  const int col = lane & 15;
  const int hi  = lane >> 4;
  const int dcol = d0 + col;
  const int dstride = HF_ * WF_;
  v8f acc = {0.f, 0.f, 0.f, 0.f, 0.f, 0.f, 0.f, 0.f};
  for (int k0 = 0; k0 < TOPK_; k0 += 4) {
    int ka = k0 + 2*hi;
    int kb = ka + 1;
    v2f a, bv;
    a.x = sW[ka];
    a.y = sW[kb];
    int doff = dcol * dstride;
    bv.x = sCoef[ka][0]*features[sBase[ka][0] + doff]
         + sCoef[ka][1]*features[sBase[ka][1] + doff]
         + sCoef[ka][2]*features[sBase[ka][2] + doff]
         + sCoef[ka][3]*features[sBase[ka][3] + doff];
    bv.y = sCoef[kb][0]*features[sBase[kb][0] + doff]
         + sCoef[kb][1]*features[sBase[kb][1] + doff]
         + sCoef[kb][2]*features[sBase[kb][2] + doff]
         + sCoef[kb][3]*features[sBase[kb][3] + doff];
    acc = __builtin_amdgcn_wmma_f32_16x16x4_f32(
        /*neg_a=*/false, a, /*neg_b=*/false, bv,
        /*c_mod=*/(short)0, acc, /*reuse_a=*/false, /*reuse_b=*/false);
  }
  if (lane < 16) sProto[d0 + lane] = acc[0];   // row M=0 of D, N=lane
  __syncthreads();

  if (tid == 0) {                               // fixed-order => deterministic
    float wsum = 0.f;
    for (int k = 0; k < TOPK_; ++k) wsum += sW[k];
    sScalar[0] = wsum;
  }
  __syncthreads();
  float wsum = sScalar[0];
  float invW = 1.f / fmaxf(wsum, EPS_);
  if (tid < DIM_) sProto[tid] *= invW;
  __syncthreads();
  if (tid == 0) {
    float ss = 0.f;
    for (int d = 0; d < DIM_; ++d) ss += sProto[d] * sProto[d];
    sScalar[1] = 1.f / fmaxf(sqrtf(ss), EPS_);
  }
  __syncthreads();
  bool has = (cls_nsel[c] > 0) && (wsum > 0.f);
  float uc = update_count[c];
  float gamma = (uc == 0.f) ? 0.f : fminf(1.f - 1.f/(uc + 1.f), GAMMA_);
  if (tid < DIM_) {
    float p    = sProto[tid] * sScalar[1];
    float oldv = prototypes[c*DIM_ + tid];
    sProto[tid] = has ? (gamma*oldv + (1.f - gamma)*p) : oldv;
  }
  __syncthreads();
  if (tid == 0) {
    float ss = 0.f;
    for (int d = 0; d < DIM_; ++d) ss += sProto[d] * sProto[d];
    sScalar[2] = 1.f / fmaxf(sqrtf(ss), EPS_);
  }
  __syncthreads();
  if (tid < DIM_) out[c*DIM_ + tid] = sProto[tid] * sScalar[2];
  if (tid == 0)  out[C_*DIM_ + c] = uc + (has ? 1.f : 0.f);
}

// ---------------------------------------------------------------------------
extern "C" void kernel_launch(void* const* d_in, const int* in_sizes, int n_in,
                              void* d_out, int out_size, void* d_ws, size_t ws_size,
                              hipStream_t stream) {
  (void)in_sizes; (void)n_in; (void)out_size; (void)ws_size;
  const float* features   = (const float*)d_in[0];
  const float* weight     = (const float*)d_in[1];
  const float* prototypes = (const float*)d_in[2];
  const float* update_cnt = (const float*)d_in[3];
  const int*   labels     = (const int*)d_in[4];
  float* out = (float*)d_out;

  char* ws = (char*)d_ws;
  float*    w_arr    = (float*)ws;                         // N_ floats (2 MB)
  uint32_t* cls_thr  = (uint32_t*)(ws + (size_t)N_ * 4);
  uint32_t* cls_need = cls_thr + C_;
  uint32_t* cls_nsel = cls_need + C_;
  uint32_t* sel_idx  = cls_nsel + C_;                      // C_*TOPK_
  float*    sel_w    = (float*)(sel_idx + C_ * TOPK_);     // C_*TOPK_

  k_weight_up<<<N_/256, 256, 0, stream>>>(weight, labels, w_arr);

  size_t ldsBytes = (size_t)(HB1 + 1024) * sizeof(uint32_t);   // 260 KB < 320 KB/WGP
  k_class_threshold<<<C_, 1024, ldsBytes, stream>>>(w_arr, labels,
                                                    cls_thr, cls_need, cls_nsel);

  k_select<<<C_, 256, 0, stream>>>(w_arr, labels, cls_thr, cls_need,
                                   sel_idx, sel_w);

  k_proto<<<C_, 512, 0, stream>>>(features, sel_idx, sel_w, cls_nsel,
                                  prototypes, update_cnt, out);
}